// QuaildGraphCutLoss_15513421873437
// MI455X (gfx1250) — compile-verified
//
#include <hip/hip_runtime.h>

// ---------------------------------------------------------------------------
// loss = max( 2 * mean(a @ b^T), 1e-7 )
//      = max( 2^-27 * dot(colsum(a), colsum(b)), 1e-7 )   [N*M = 2^28]
//
// Stage 1: deterministic partial column sums (HBM-bandwidth bound, b128 loads)
// Stage 2: final reduction; dot product summed on the WMMA pipe
//          (A = ones 16x4 f32  =>  D[m][n] = sum_k B[k][n] + C[m][n])
// ---------------------------------------------------------------------------

typedef __attribute__((ext_vector_type(2))) float v2f;
typedef __attribute__((ext_vector_type(8))) float v8f;

constexpr int N_ROWS = 16384;   // rows of a (== rows of b)
constexpr int K_DIM  = 512;     // shared inner dim
constexpr int K4     = K_DIM / 4;

// ---- Stage 1: each block sums `rows_per_block` rows of one input matrix.
// blockDim.x = 128 threads; thread t owns float4 column group t (128*4 = 512).
// blockIdx.y selects a (0) or b (1). Output: one 512-float partial row per
// block, fully written (no init / no atomics -> bitwise deterministic).
__global__ __launch_bounds__(128) void colsum_partial_kernel(
    const float* __restrict__ a, const float* __restrict__ b,
    float* __restrict__ partials, int rows_per_block)
{
    const int t     = threadIdx.x;     // 0..127
    const int blk   = blockIdx.x;
    const int which = blockIdx.y;      // 0 = a, 1 = b

    const float4* src = reinterpret_cast<const float4*>(which ? b : a);
    const size_t  r0  = (size_t)blk * (size_t)rows_per_block;

    float4 acc = make_float4(0.f, 0.f, 0.f, 0.f);
#pragma unroll 4
    for (int r = 0; r < rows_per_block; ++r) {
        float4 v = src[(r0 + (size_t)r) * K4 + t];   // coalesced b128 stream
        acc.x += v.x; acc.y += v.y; acc.z += v.z; acc.w += v.w;
    }

    float4* dst = reinterpret_cast<float4*>(partials);
    dst[((size_t)which * gridDim.x + blk) * K4 + t] = acc;
}

// ---- Stage 2: one block. Threads 0..511 finish colsum(a) column t,
// threads 512..1023 finish colsum(b). Wave 0 then evaluates
// dot(sA,sB) on the matrix pipe and writes the scalar loss.
__global__ __launch_bounds__(1024) void finalize_kernel(
    const float* __restrict__ partials, float* __restrict__ out, int nblocks)
{
    __shared__ float s[2 * K_DIM];     // s[0..511]=colsum(a), s[512..1023]=colsum(b)

    const int t     = threadIdx.x;     // 0..1023
    const int which = t >> 9;
    const int k     = t & (K_DIM - 1);

    const float* base = partials + ((size_t)which * (size_t)nblocks) * K_DIM + k;
    float acc = 0.f;
    for (int i = 0; i < nblocks; ++i) acc += base[(size_t)i * K_DIM];
    s[t] = acc;
    __syncthreads();

    if (t < 32) {   // wave 0 only; all 32 lanes active -> EXEC all ones for WMMA
        // Lane L holds products p[L + 32*j], j = 0..15  (covers all 512 once).
        float p[16];
#pragma unroll
        for (int j = 0; j < 16; ++j) {
            const int kk = t + 32 * j;
            p[j] = s[kk] * s[K_DIM + kk];
        }

        // A = ones(16x4): D[m][n] = sum over the 4 K-slots of B[:,n] + C[m][n].
        // Iteration i feeds the 64 products p[64i .. 64i+63] as the B tile
        // (b.x = p[64i+L], b.y = p[64i+32+L]); every product enters exactly
        // one (k,n) slot of some iteration, so chaining C accumulates
        // D[0][n] such that sum_n D[0][n] = dot(sA, sB).
        v2f ones; ones.x = 1.0f; ones.y = 1.0f;
        v8f c = {};
#pragma unroll
        for (int i = 0; i < 8; ++i) {
            v2f bb; bb.x = p[2 * i]; bb.y = p[2 * i + 1];
            c = __builtin_amdgcn_wmma_f32_16x16x4_f32(
                    /*neg_a=*/false, ones,
                    /*neg_b=*/false, bb,
                    /*c_mod=*/(short)0, c,
                    /*reuse_a=*/false, /*reuse_b=*/false);
        }

        // C/D layout: VGPR0 = row M=0 (lanes 0-15, N=lane) and row M=8
        // (lanes 16-31). All D rows are identical (ones A), so lanes 0-15
        // carry the 16 distinct column accumulators; zero the duplicates.
        float v = (t < 16) ? c[0] : 0.f;
#pragma unroll
        for (int off = 16; off >= 1; off >>= 1)
            v += __shfl_xor(v, off, 32);

        if (t == 0) {
            // 2*LAMBD / (N*M) = 2 / 2^28 = 2^-27 (exact in fp32)
            const float loss = v * (2.0f / ((float)N_ROWS * (float)N_ROWS));
            out[0] = fmaxf(loss, 1e-7f);
        }
    }
}

extern "C" void kernel_launch(void* const* d_in, const int* in_sizes, int n_in,
                              void* d_out, int out_size, void* d_ws, size_t ws_size,
                              hipStream_t stream) {
    (void)in_sizes; (void)n_in; (void)out_size;

    const float* a   = (const float*)d_in[0];
    const float* b   = (const float*)d_in[1];
    float*       out = (float*)d_out;
    float*       ws  = (float*)d_ws;

    // Partial-block count sized to scratch (deterministic for a fixed ws_size).
    int nblocks = 256;                                   // 1 MiB of partials
    while ((size_t)2 * (size_t)nblocks * K_DIM * sizeof(float) > ws_size &&
           nblocks > 1)
        nblocks >>= 1;
    const int rows_per_block = N_ROWS / nblocks;

    dim3 grid1(nblocks, 2);
    colsum_partial_kernel<<<grid1, 128, 0, stream>>>(a, b, ws, rows_per_block);
    finalize_kernel<<<1, 1024, 0, stream>>>(ws, out, nblocks);
}